// RepresentationNetwork_52338471469708
// MI455X (gfx1250) — compile-verified
//
#include <hip/hip_runtime.h>

#define Bn   512
#define Ln   256
#define Dn   64
#define Hn   256
#define TOPK 128
#define ROWS (Bn*Ln)

// workspace layout (float offsets)
#define WS_M    0
#define WS_P    4096
#define WS_Q    4160
#define WS_C    4224
#define WS_T    4352
#define WS_U    (WS_T + ROWS*Dn)
#define WS_V    (WS_U + ROWS)
#define WS_PAD  (WS_V + ROWS)
#define WS_POOL (WS_PAD + ROWS)
#define WS_H1   (WS_POOL + Bn*Hn)

typedef float v2f __attribute__((ext_vector_type(2)));
typedef float v8f __attribute__((ext_vector_type(8)));

static __device__ __forceinline__ v8f wmma4(v2f a, v2f b, v8f c) {
  // V_WMMA_F32_16X16X4_F32 : D = A(16x4) * B(4x16) + C
  return __builtin_amdgcn_wmma_f32_16x16x4_f32(false, a, false, b, (short)0, c,
                                               false, false);
}

static __device__ __forceinline__ unsigned okey(float s) {
  unsigned u = __float_as_uint(s);
  return (u & 0x80000000u) ? ~u : (u | 0x80000000u);
}

// ---------------- K0: M = Wq*Wk^T, p = Wq*bk, q = Wk*bq, c = bq.bk ----------
__global__ void k_prep(const float* __restrict__ Wq, const float* __restrict__ bq,
                       const float* __restrict__ Wk, const float* __restrict__ bk,
                       float* __restrict__ ws) {
  int t = threadIdx.x;  // 256 threads, 1 block
  for (int e = t; e < Dn * Dn; e += 256) {
    int d0 = e >> 6, d1 = e & 63;
    float acc = 0.f;
    for (int h = 0; h < Hn; ++h) acc += Wq[d0 * Hn + h] * Wk[d1 * Hn + h];
    ws[WS_M + e] = acc;
  }
  if (t < Dn) {
    float a = 0.f, b = 0.f;
    for (int h = 0; h < Hn; ++h) {
      a += Wq[t * Hn + h] * bk[h];
      b += Wk[t * Hn + h] * bq[h];
    }
    ws[WS_P + t] = a;
    ws[WS_Q + t] = b;
  }
  if (t == 0) {
    float c = 0.f;
    for (int h = 0; h < Hn; ++h) c += bq[h] * bk[h];
    ws[WS_C] = c;
  }
}

// ---------------- K0b: T = x @ M  (one wave per 16x16 tile, K=64) ----------
__global__ void k_xT(const float* __restrict__ x, float* __restrict__ ws) {
  int g0 = blockIdx.x * 16;   // row tile over B*L
  int d0 = blockIdx.y * 16;   // col tile over 64
  int lane = threadIdx.x;
  int nn = lane & 15, kh = (lane >> 4) * 2;
  const float* Ar = x + (size_t)(g0 + nn) * Dn;
  const float* Mm = ws + WS_M;
  v8f acc = {};
  for (int kt = 0; kt < 16; ++kt) {
    int k = kt * 4 + kh;
    v2f a = {Ar[k], Ar[k + 1]};
    v2f b = {Mm[k * Dn + d0 + nn], Mm[(k + 1) * Dn + d0 + nn]};
    acc = wmma4(a, b, acc);
  }
  int rb = g0 + ((lane >> 4) << 3);
  for (int i = 0; i < 8; ++i)
    ws[WS_T + (size_t)(rb + i) * Dn + d0 + nn] = acc[i];
}

// ---------------- K0c: u = x.p, v = x.q, pad = (sum|x| != 0) ----------------
__global__ void k_uvp(const float* __restrict__ x, float* __restrict__ ws) {
  int g = blockIdx.x * blockDim.x + threadIdx.x;
  if (g >= ROWS) return;
  const float* xr = x + (size_t)g * Dn;
  const float* pv = ws + WS_P;
  const float* qv = ws + WS_Q;
  float uu = 0.f, vv = 0.f, aa = 0.f;
  for (int d = 0; d < Dn; ++d) {
    float xv = xr[d];
    uu += xv * pv[d];
    vv += xv * qv[d];
    aa += fabsf(xv);
  }
  ws[WS_U + g] = uu;
  ws[WS_V + g] = vv;
  ws[WS_PAD + g] = (aa != 0.f) ? 1.f : 0.f;
}

// score tile:  S[l0+i, m0+n] = (T x^T + u_l + v_m + c)/16, masked by pad
static __device__ __forceinline__ void score_tile(
    const float* __restrict__ Tb, const float* __restrict__ xb,
    const float* __restrict__ ub, const float* __restrict__ vb,
    const float* __restrict__ pb, float cc, int l0, int m0, int lane,
    float s[8]) {
  int nn = lane & 15, kh = (lane >> 4) * 2;
  const float* Ar = Tb + (size_t)(l0 + nn) * Dn;
  const float* Br = xb + (size_t)(m0 + nn) * Dn;
  v8f acc = {};
  for (int kt = 0; kt < 16; ++kt) {
    int k = kt * 4 + kh;
    v2f a = {Ar[k], Ar[k + 1]};
    v2f b = {Br[k], Br[k + 1]};  // B[k][n] = x[m0+n][k] -> contiguous pair
    acc = wmma4(a, b, acc);
  }
  int m = m0 + nn;
  float vm = vb[m], pm = pb[m];
  int rb = l0 + ((lane >> 4) << 3);
  for (int i = 0; i < 8; ++i) {
    int l = rb + i;
    float sv = (acc[i] + ub[l] + vm + cc) * 0.0625f;
    if (pb[l] == 0.f || pm == 0.f) sv = -__builtin_inff();
    s[i] = sv;
  }
}

// ---------------- K1: per-batch scores + radix top-128 + softmax + MLPs ----
__global__ __launch_bounds__(512) void k_main(
    const float* __restrict__ x, float* __restrict__ ws,
    const float* __restrict__ phi_w1, const float* __restrict__ phi_b1,
    const float* __restrict__ phi_w2, const float* __restrict__ phi_b2,
    const float* __restrict__ xi_w1, const float* __restrict__ xi_b1,
    const float* __restrict__ xi_w2, const float* __restrict__ xi_b2) {
  __shared__ float sh_w[TOPK];
  __shared__ int sh_idx[TOPK];
  __shared__ unsigned sh_hist[256];
  __shared__ unsigned sh_prefix;
  __shared__ int sh_rem, sh_nab, sh_ntie;
  __shared__ float sh_red[TOPK];
  __shared__ float sh_scal, sh_selfsum;
  __shared__ float sh_ps[Hn], sh_pp[Hn];

  int b = blockIdx.x;
  int tid = threadIdx.x;
  int lane = tid & 31, wave = tid >> 5;  // 16 wave32s
  const float* xb = x + (size_t)b * Ln * Dn;
  const float* Tb = ws + WS_T + (size_t)b * Ln * Dn;
  const float* ub = ws + WS_U + b * Ln;
  const float* vb = ws + WS_V + b * Ln;
  const float* pb = ws + WS_PAD + b * Ln;
  float cc = ws[WS_C];

  if (tid == 0) { sh_prefix = 0u; sh_rem = TOPK; }

  // ---- 4-pass MSB radix select: exact 128th-largest key ----
  for (int p = 0; p < 4; ++p) {
    if (tid < 256) sh_hist[tid] = 0u;
    __syncthreads();
    unsigned prefix = sh_prefix;
    for (int t = wave; t < 256; t += 16) {
      int l0 = (t >> 4) << 4, m0 = (t & 15) << 4;
      float s[8];
      score_tile(Tb, xb, ub, vb, pb, cc, l0, m0, lane, s);
      for (int i = 0; i < 8; ++i) {
        unsigned k = okey(s[i]);
        bool match = (p == 0) || ((k >> (32 - 8 * p)) == prefix);
        if (match) atomicAdd(&sh_hist[(k >> (24 - 8 * p)) & 255u], 1u);
      }
    }
    __syncthreads();
    if (tid == 0) {
      int rem = sh_rem;
      unsigned cum = 0;
      int chosen = 0;
      for (int bk2 = 255; bk2 >= 0; --bk2) {
        unsigned c2 = cum + sh_hist[bk2];
        if ((int)c2 >= rem) { chosen = bk2; sh_rem = rem - (int)cum; break; }
        cum = c2;
      }
      sh_prefix = (sh_prefix << 8) | (unsigned)chosen;
    }
    __syncthreads();
  }
  unsigned thr = sh_prefix;
  int tieneed = sh_rem;
  int nabove = TOPK - tieneed;
  if (tid == 0) { sh_nab = 0; sh_ntie = 0; }
  __syncthreads();

  // ---- collection pass (set is permutation-invariant downstream) ----
  for (int t = wave; t < 256; t += 16) {
    int l0 = (t >> 4) << 4, m0 = (t & 15) << 4;
    float s[8];
    score_tile(Tb, xb, ub, vb, pb, cc, l0, m0, lane, s);
    int m = m0 + (lane & 15);
    int rb = l0 + ((lane >> 4) << 3);
    for (int i = 0; i < 8; ++i) {
      unsigned k = okey(s[i]);
      if (k > thr) {
        int slot = atomicAdd(&sh_nab, 1);
        if (slot < nabove) { sh_w[slot] = s[i]; sh_idx[slot] = ((rb + i) << 8) | m; }
      } else if (k == thr) {
        int slot = atomicAdd(&sh_ntie, 1);
        if (slot < tieneed) {
          sh_w[nabove + slot] = s[i];
          sh_idx[nabove + slot] = ((rb + i) << 8) | m;
        }
      }
    }
  }
  __syncthreads();

  // ---- softmax over the 128 selected values ----
  if (tid < TOPK) sh_red[tid] = sh_w[tid];
  __syncthreads();
  for (int off = 64; off > 0; off >>= 1) {
    if (tid < off) sh_red[tid] = fmaxf(sh_red[tid], sh_red[tid + off]);
    __syncthreads();
  }
  if (tid == 0) sh_scal = sh_red[0];
  __syncthreads();
  float mx = sh_scal;
  if (tid < TOPK) { float e = __expf(sh_w[tid] - mx); sh_w[tid] = e; sh_red[tid] = e; }
  __syncthreads();
  for (int off = 64; off > 0; off >>= 1) {
    if (tid < off) sh_red[tid] += sh_red[tid + off];
    __syncthreads();
  }
  if (tid == 0) { sh_scal = sh_red[0]; sh_selfsum = 0.f; }
  __syncthreads();
  if (tid < TOPK) sh_w[tid] /= sh_scal;
  if (tid < Hn) { sh_ps[tid] = 0.f; sh_pp[tid] = 0.f; }
  __syncthreads();
  if (tid < TOPK) {
    int id = sh_idx[tid];
    if ((id >> 8) == (id & 255)) atomicAdd(&sh_selfsum, sh_w[tid]);
  }
  __syncthreads();

  // ---- layer-1 MLPs on gathered rows (WMMA), weighted-pool the relu acts --
  {
    int nn = lane & 15, kh = (lane >> 4) * 2;
    for (int j = wave; j < 256; j += 16) {
      int type = j >> 7;  // 0: phi (self), 1: xi (pair)
      int mt = (j >> 4) & 7, nt = j & 15;
      int id = sh_idx[mt * 16 + nn];
      int r = id >> 8, c = id & 255;
      v8f acc = {};
      if (type == 0) {
        const float* Ar = xb + (size_t)r * Dn;
        const float* Bc = phi_w1 + nt * 16 + nn;
        for (int kt = 0; kt < 16; ++kt) {
          int k = kt * 4 + kh;
          v2f a = {Ar[k], Ar[k + 1]};
          v2f bv = {Bc[k * Hn], Bc[(k + 1) * Hn]};
          acc = wmma4(a, bv, acc);
        }
      } else {
        const float* Ar0 = xb + (size_t)r * Dn;
        const float* Ar1 = xb + (size_t)c * Dn;
        const float* Bc = xi_w1 + nt * 16 + nn;
        for (int kt = 0; kt < 32; ++kt) {
          int k = kt * 4 + kh;  // both k,k+1 on same side of 64
          float a0, a1;
          if (k < Dn) { a0 = Ar0[k]; a1 = Ar0[k + 1]; }
          else        { a0 = Ar1[k - Dn]; a1 = Ar1[k - Dn + 1]; }
          v2f a = {a0, a1};
          v2f bv = {Bc[k * Hn], Bc[(k + 1) * Hn]};
          acc = wmma4(a, bv, acc);
        }
      }
      int n = nt * 16 + nn;
      float bias = (type == 0) ? phi_b1[n] : xi_b1[n];
      int kb = mt * 16 + ((lane >> 4) << 3);
      for (int i = 0; i < 8; ++i) {
        int kk = kb + i;
        int id2 = sh_idx[kk];
        bool isself = ((id2 >> 8) == (id2 & 255));
        float act = fmaxf(acc[i] + bias, 0.f);
        float wgt = sh_w[kk];
        if (type == 0) { if (isself)  atomicAdd(&sh_ps[n], wgt * act); }
        else           { if (!isself) atomicAdd(&sh_pp[n], wgt * act); }
      }
    }
  }
  __syncthreads();

  // ---- pooled = ps @ phi_w2 + pp @ xi_w2 + blended layer-2 biases ----
  if (tid < Hn) {
    float ss = sh_selfsum;
    float acc = ss * phi_b2[tid] + (1.f - ss) * xi_b2[tid];
    for (int h = 0; h < Hn; ++h)
      acc += sh_ps[h] * phi_w2[h * Hn + tid] + sh_pp[h] * xi_w2[h * Hn + tid];
    ws[WS_POOL + b * Hn + tid] = acc;
  }
}

// ---------------- K2a: H1 = relu(pooled @ rho_w1 + b1) ----------------
__global__ void k_rho1(float* __restrict__ ws, const float* __restrict__ rho_w1,
                       const float* __restrict__ rho_b1) {
  int bt = blockIdx.x, nt = blockIdx.y;
  int lane = threadIdx.x;
  int nn = lane & 15, kh = (lane >> 4) * 2;
  const float* Ar = ws + WS_POOL + (size_t)(bt * 16 + nn) * Hn;
  const float* Bc = rho_w1 + nt * 16 + nn;
  v8f acc = {};
  for (int kt = 0; kt < 64; ++kt) {
    int k = kt * 4 + kh;
    v2f a = {Ar[k], Ar[k + 1]};
    v2f bv = {Bc[k * Hn], Bc[(k + 1) * Hn]};
    acc = wmma4(a, bv, acc);
  }
  int n = nt * 16 + nn;
  float bias = rho_b1[n];
  int rb = bt * 16 + ((lane >> 4) << 3);
  for (int i = 0; i < 8; ++i)
    ws[WS_H1 + (size_t)(rb + i) * Hn + n] = fmaxf(acc[i] + bias, 0.f);
}

// ---------------- K2b: out = H1 @ rho_w2 + b2 ----------------
__global__ void k_rho2(const float* __restrict__ ws,
                       const float* __restrict__ rho_w2,
                       const float* __restrict__ rho_b2,
                       float* __restrict__ out) {
  int bt = blockIdx.x, nt = blockIdx.y;  // nt over 128/16
  int lane = threadIdx.x;
  int nn = lane & 15, kh = (lane >> 4) * 2;
  const float* Ar = ws + WS_H1 + (size_t)(bt * 16 + nn) * Hn;
  const float* Bc = rho_w2 + nt * 16 + nn;  // [256,128] row-major
  v8f acc = {};
  for (int kt = 0; kt < 64; ++kt) {
    int k = kt * 4 + kh;
    v2f a = {Ar[k], Ar[k + 1]};
    v2f bv = {Bc[k * 128], Bc[(k + 1) * 128]};
    acc = wmma4(a, bv, acc);
  }
  int n = nt * 16 + nn;
  float bias = rho_b2[n];
  int rb = bt * 16 + ((lane >> 4) << 3);
  for (int i = 0; i < 8; ++i) out[(size_t)(rb + i) * 128 + n] = acc[i] + bias;
}

extern "C" void kernel_launch(void* const* d_in, const int* in_sizes, int n_in,
                              void* d_out, int out_size, void* d_ws,
                              size_t ws_size, hipStream_t stream) {
  (void)in_sizes; (void)n_in; (void)out_size; (void)ws_size;
  const float* x      = (const float*)d_in[0];
  const float* Wq     = (const float*)d_in[1];
  const float* bq     = (const float*)d_in[2];
  const float* Wk     = (const float*)d_in[3];
  const float* bk     = (const float*)d_in[4];
  const float* phi_w1 = (const float*)d_in[5];
  const float* phi_b1 = (const float*)d_in[6];
  const float* phi_w2 = (const float*)d_in[7];
  const float* phi_b2 = (const float*)d_in[8];
  const float* xi_w1  = (const float*)d_in[9];
  const float* xi_b1  = (const float*)d_in[10];
  const float* xi_w2  = (const float*)d_in[11];
  const float* xi_b2  = (const float*)d_in[12];
  const float* rho_w1 = (const float*)d_in[13];
  const float* rho_b1 = (const float*)d_in[14];
  const float* rho_w2 = (const float*)d_in[15];
  const float* rho_b2 = (const float*)d_in[16];
  float* ws  = (float*)d_ws;
  float* out = (float*)d_out;

  k_prep<<<1, 256, 0, stream>>>(Wq, bq, Wk, bk, ws);
  k_xT<<<dim3(ROWS / 16, Dn / 16), 32, 0, stream>>>(x, ws);
  k_uvp<<<ROWS / 256, 256, 0, stream>>>(x, ws);
  k_main<<<Bn, 512, 0, stream>>>(x, ws, phi_w1, phi_b1, phi_w2, phi_b2, xi_w1,
                                 xi_b1, xi_w2, xi_b2);
  k_rho1<<<dim3(Bn / 16, Hn / 16), 32, 0, stream>>>(ws, rho_w1, rho_b1);
  k_rho2<<<dim3(Bn / 16, 128 / 16), 32, 0, stream>>>(ws, rho_w2, rho_b2, out);
}